// SelfSeqAtten_3109556322512
// MI455X (gfx1250) — compile-verified
//
#include <hip/hip_runtime.h>
#include <math.h>

typedef float v2f __attribute__((ext_vector_type(2)));
typedef float v8f __attribute__((ext_vector_type(8)));
typedef unsigned int tdm_v4u __attribute__((ext_vector_type(4)));
typedef int tdm_v8i __attribute__((ext_vector_type(8)));
typedef int tdm_v4i __attribute__((ext_vector_type(4)));

constexpr int kS = 1024;
constexpr int kB = 32;
constexpr int kD = 512;
constexpr int kKT = 32;      // K tile depth (floats)
constexpr int kSTR = 36;     // padded LDS row stride (32 data + 4 pad dwords)
constexpr int kGSTR = 264;   // padded LDS row stride (256 data + 8 pad dwords)
constexpr float kEPS = 1e-6f;

// ---------------------------------------------------------------------------
// Issue one TDM 2-D tile load: tile_d0 (contiguous dwords) x tile_d1 rows,
// row stride = stride_elems (dwords) in global memory; LDS rows padded by
// pad_interval/pad_amount (hardware padding reproduces our LDS strides).
// D# layout per CDNA5 ISA ch.8 (trailing groups zero => 2-D tensor).
// Toolchain uses the 6-arg builtin: (v4u, v8i, v4i, v4i, v8i, i32 cpol).
// ---------------------------------------------------------------------------
__device__ __forceinline__ void tdm_load_2d(unsigned lds_off, const void* gaddr,
                                            unsigned tile_d0, unsigned tile_d1,
                                            unsigned long long stride_elems,
                                            unsigned pad_interval, unsigned pad_amount) {
  const unsigned long long ga = (unsigned long long)gaddr;
  tdm_v4u g0;
  g0.x = 1u;                                             // count=1, user mode
  g0.y = lds_off;                                        // lds_addr [63:32]
  g0.z = (unsigned)ga;                                   // global_addr[31:0]
  g0.w = (unsigned)((ga >> 32) & 0x1FFFFFFull) | (2u << 30); // ga[56:32] | type=2
  tdm_v8i g1;
  g1[0] = (int)((2u << 16) | (1u << 20) | (pad_interval << 22) | (pad_amount << 25));
  g1[1] = (int)((tile_d0 & 0xFFFFu) << 16);                         // tensor_dim0 lo
  g1[2] = (int)(((tile_d0 >> 16) & 0xFFFFu) | ((tile_d1 & 0xFFFFu) << 16)); // td0 hi | td1 lo
  g1[3] = (int)(((tile_d1 >> 16) & 0xFFFFu) | ((tile_d0 & 0xFFFFu) << 16)); // td1 hi | tile_dim0
  g1[4] = (int)(tile_d1 & 0xFFFFu);                                 // tile_dim1 (tile_dim2=0)
  g1[5] = (int)(unsigned)(stride_elems & 0xFFFFFFFFull);            // dim0_stride[31:0]
  g1[6] = (int)(unsigned)((stride_elems >> 32) & 0xFFFFull);        // dim0_stride[47:32]
  g1[7] = 0;
  const tdm_v4i z4 = (tdm_v4i)0;
  const tdm_v8i z8 = (tdm_v8i)0;
  __builtin_amdgcn_tensor_load_to_lds(g0, g1, z4, z4, z8, 0);
}

// ---------------------------------------------------------------------------
// Kernel 1: x[b,s,t] = mask[b,t] * (h[s,b,:] . h[t,b,:]),  x[b,s,s] = 0
// TDM double-buffered LDS pipeline; wave tile 64x64, WG tile 128x256.
// ---------------------------------------------------------------------------
__global__ __launch_bounds__(256) void scores_kernel(const float* __restrict__ h,
                                                     const int* __restrict__ mask,
                                                     float* __restrict__ xbuf) {
  __shared__ __align__(16) float As[2 * 128 * kSTR];
  __shared__ __align__(16) float Bs[2 * 256 * kSTR];
  __shared__ float Ms[256];

  const int mblk = blockIdx.x, nblk = blockIdx.y, b = blockIdx.z;
  const int tid = threadIdx.x;
  const int lane = tid & 31, wid = tid >> 5;
  const int wm = wid & 1, wn = wid >> 1;        // 2 x 4 wave grid
  const int hl = lane >> 4, l15 = lane & 15;

  Ms[tid] = (float)mask[b * kS + nblk * 256 + tid];

  const float* hA = h + (size_t)(mblk * 128) * (kB * kD) + b * kD;
  const float* hB = h + (size_t)(nblk * 256) * (kB * kD) + b * kD;
  const unsigned asOff = (unsigned)(size_t)(void*)As;   // flat->LDS: low 32 bits = LDS offset
  const unsigned bsOff = (unsigned)(size_t)(void*)Bs;
  constexpr unsigned kABUF = 128 * kSTR * 4;
  constexpr unsigned kBBUF = 256 * kSTR * 4;

  v8f acc[4][4];
#pragma unroll
  for (int i = 0; i < 4; ++i)
#pragma unroll
    for (int j = 0; j < 4; ++j) acc[i][j] = (v8f)0.0f;

  constexpr int T = kD / kKT;   // 16
  if (wid == 0) {
    tdm_load_2d(asOff, hA, kKT, 128, (unsigned long long)(kB * kD), 4, 3);
    tdm_load_2d(bsOff, hB, kKT, 256, (unsigned long long)(kB * kD), 4, 3);
  }
  for (int t = 0; t < T; ++t) {
    if (wid == 0) {
      if (t + 1 < T) {
        const unsigned bn = (unsigned)(t + 1) & 1u;
        tdm_load_2d(asOff + bn * kABUF, hA + (t + 1) * kKT, kKT, 128,
                    (unsigned long long)(kB * kD), 4, 3);
        tdm_load_2d(bsOff + bn * kBBUF, hB + (t + 1) * kKT, kKT, 256,
                    (unsigned long long)(kB * kD), 4, 3);
        __builtin_amdgcn_s_wait_tensorcnt(2);   // oldest pair (this slab) complete
      } else {
        __builtin_amdgcn_s_wait_tensorcnt(0);
      }
    }
    __syncthreads();
    const float* Ab = As + (t & 1) * (128 * kSTR);
    const float* Bb = Bs + (t & 1) * (256 * kSTR);
#pragma unroll
    for (int kk = 0; kk < kKT; kk += 4) {
      v2f af[4], bf[4];
#pragma unroll
      for (int i = 0; i < 4; ++i)
        af[i] = *(const v2f*)(Ab + (wm * 64 + i * 16 + l15) * kSTR + kk + 2 * hl);
#pragma unroll
      for (int j = 0; j < 4; ++j)
        bf[j] = *(const v2f*)(Bb + (wn * 64 + j * 16 + l15) * kSTR + kk + 2 * hl);
#pragma unroll
      for (int i = 0; i < 4; ++i)
#pragma unroll
        for (int j = 0; j < 4; ++j)
          acc[i][j] = __builtin_amdgcn_wmma_f32_16x16x4_f32(
              false, af[i], false, bf[j], (short)0, acc[i][j], false, false);
    }
    __syncthreads();   // all waves done with buffer (t&1) before TDM refills it
  }

  const int m0 = mblk * 128 + wm * 64;
  const int n0 = nblk * 256 + wn * 64;
#pragma unroll
  for (int i = 0; i < 4; ++i)
#pragma unroll
    for (int j = 0; j < 4; ++j) {
      const int gn = n0 + j * 16 + l15;
      const float mv = Ms[wn * 64 + j * 16 + l15];
#pragma unroll
      for (int g = 0; g < 8; ++g) {
        const int gm = m0 + i * 16 + hl * 8 + g;
        float x = acc[i][j][g] * mv;
        if (gm == gn) x = 0.0f;
        xbuf[(size_t)b * kS * kS + (size_t)gm * kS + gn] = x;
      }
    }
}

// ---------------------------------------------------------------------------
// Kernel 2: in-place masked softmax over each row of alpha (one wave per row).
// ---------------------------------------------------------------------------
__global__ __launch_bounds__(256) void softmax_kernel(const int* __restrict__ mask,
                                                      float* __restrict__ alpha) {
  const int lane = threadIdx.x & 31;
  const int wid = threadIdx.x >> 5;
  const int rowIdx = blockIdx.x * 8 + wid;      // 0 .. B*S-1
  const int b = rowIdx >> 10;

  float* rp = alpha + (size_t)rowIdx * kS;
  const int* mp = mask + (size_t)b * kS;

  float4 xv[8];
  float mx = -INFINITY;
#pragma unroll
  for (int i = 0; i < 8; ++i) {
    xv[i] = *(const float4*)(rp + i * 128 + lane * 4);
    mx = fmaxf(mx, fmaxf(fmaxf(xv[i].x, xv[i].y), fmaxf(xv[i].z, xv[i].w)));
  }
#pragma unroll
  for (int off = 16; off > 0; off >>= 1) mx = fmaxf(mx, __shfl_xor(mx, off, 32));

  float sum = 0.0f;
#pragma unroll
  for (int i = 0; i < 8; ++i) {
    int4 m4 = *(const int4*)(mp + i * 128 + lane * 4);
    float4 e;
    e.x = expf(xv[i].x - mx) * (float)m4.x;
    e.y = expf(xv[i].y - mx) * (float)m4.y;
    e.z = expf(xv[i].z - mx) * (float)m4.z;
    e.w = expf(xv[i].w - mx) * (float)m4.w;
    sum += (e.x + e.y) + (e.z + e.w);
    xv[i] = e;
  }
#pragma unroll
  for (int off = 16; off > 0; off >>= 1) sum += __shfl_xor(sum, off, 32);

  const float inv = 1.0f / (sum + kEPS);
#pragma unroll
  for (int i = 0; i < 8; ++i) {
    float4 o;
    o.x = xv[i].x * inv; o.y = xv[i].y * inv;
    o.z = xv[i].z * inv; o.w = xv[i].w * inv;
    *(float4*)(rp + i * 128 + lane * 4) = o;
  }
}

// ---------------------------------------------------------------------------
// Kernel 3: out[s,b,d] = sum_t alpha[b,s,t] * h[t,b,d]
// alpha tile via TDM (k-contiguous); h tile via TDM landing [k][d]-major,
// B-fragments read as two b32 LDS loads (stride 264 -> conflict-free halves).
// ---------------------------------------------------------------------------
__global__ __launch_bounds__(256) void out_kernel(const float* __restrict__ h,
                                                  const float* __restrict__ alpha,
                                                  float* __restrict__ out) {
  __shared__ __align__(16) float As[2 * 128 * kSTR];
  __shared__ __align__(16) float Gs[2 * kKT * kGSTR];

  const int mblk = blockIdx.x, nblk = blockIdx.y, b = blockIdx.z;
  const int tid = threadIdx.x;
  const int lane = tid & 31, wid = tid >> 5;
  const int wm = wid & 1, wn = wid >> 1;
  const int hl = lane >> 4, l15 = lane & 15;

  const float* aA = alpha + (size_t)b * kS * kS + (size_t)(mblk * 128) * kS;
  const float* hG = h + (size_t)b * kD + (size_t)nblk * 256;
  const unsigned asOff = (unsigned)(size_t)(void*)As;
  const unsigned gsOff = (unsigned)(size_t)(void*)Gs;
  constexpr unsigned kABUF = 128 * kSTR * 4;
  constexpr unsigned kGBUF = kKT * kGSTR * 4;

  v8f acc[4][4];
#pragma unroll
  for (int i = 0; i < 4; ++i)
#pragma unroll
    for (int j = 0; j < 4; ++j) acc[i][j] = (v8f)0.0f;

  constexpr int T = kS / kKT;   // 32
  if (wid == 0) {
    tdm_load_2d(asOff, aA, kKT, 128, (unsigned long long)kS, 4, 3);
    tdm_load_2d(gsOff, hG, 256, kKT, (unsigned long long)(kB * kD), 7, 7);
  }
  for (int t = 0; t < T; ++t) {
    if (wid == 0) {
      if (t + 1 < T) {
        const unsigned bn = (unsigned)(t + 1) & 1u;
        tdm_load_2d(asOff + bn * kABUF, aA + (t + 1) * kKT, kKT, 128,
                    (unsigned long long)kS, 4, 3);
        tdm_load_2d(gsOff + bn * kGBUF,
                    hG + (size_t)(t + 1) * kKT * (kB * kD), 256, kKT,
                    (unsigned long long)(kB * kD), 7, 7);
        __builtin_amdgcn_s_wait_tensorcnt(2);
      } else {
        __builtin_amdgcn_s_wait_tensorcnt(0);
      }
    }
    __syncthreads();
    const float* Ab = As + (t & 1) * (128 * kSTR);
    const float* Gb = Gs + (t & 1) * (kKT * kGSTR);
#pragma unroll
    for (int kk = 0; kk < kKT; kk += 4) {
      v2f af[4], bf[4];
#pragma unroll
      for (int i = 0; i < 4; ++i)
        af[i] = *(const v2f*)(Ab + (wm * 64 + i * 16 + l15) * kSTR + kk + 2 * hl);
      const int r = kk + 2 * hl;
#pragma unroll
      for (int j = 0; j < 4; ++j) {
        const int c = wn * 64 + j * 16 + l15;
        bf[j].x = Gb[(r + 0) * kGSTR + c];
        bf[j].y = Gb[(r + 1) * kGSTR + c];
      }
#pragma unroll
      for (int i = 0; i < 4; ++i)
#pragma unroll
        for (int j = 0; j < 4; ++j)
          acc[i][j] = __builtin_amdgcn_wmma_f32_16x16x4_f32(
              false, af[i], false, bf[j], (short)0, acc[i][j], false, false);
    }
    __syncthreads();
  }

  const int m0 = mblk * 128 + wm * 64;
  const int n0 = nblk * 256 + wn * 64;
#pragma unroll
  for (int i = 0; i < 4; ++i)
#pragma unroll
    for (int j = 0; j < 4; ++j) {
      const int gd = n0 + j * 16 + l15;
#pragma unroll
      for (int g = 0; g < 8; ++g) {
        const int gm = m0 + i * 16 + hl * 8 + g;
        out[(size_t)gm * (kB * kD) + b * kD + gd] = acc[i][j][g];
      }
    }
}

// ---------------------------------------------------------------------------
extern "C" void kernel_launch(void* const* d_in, const int* in_sizes, int n_in,
                              void* d_out, int out_size, void* d_ws, size_t ws_size,
                              hipStream_t stream) {
  (void)in_sizes; (void)n_in; (void)out_size; (void)d_ws; (void)ws_size;
  const float* h = (const float*)d_in[0];
  const int* h_mask = (const int*)d_in[1];
  float* out = (float*)d_out;
  float* alpha = out + (size_t)kS * kB * kD;   // outputs: [out | alpha]

  scores_kernel<<<dim3(kS / 128, kS / 256, kB), 256, 0, stream>>>(h, h_mask, alpha);
  softmax_kernel<<<dim3((kB * kS) / 8), 256, 0, stream>>>(h_mask, alpha);
  out_kernel<<<dim3(kS / 128, kD / 256, kB), 256, 0, stream>>>(h, alpha, out);
}